// interactionModule_42769284333963
// MI455X (gfx1250) — compile-verified
//
#include <hip/hip_runtime.h>
#include <stdint.h>

// ---------------------------------------------------------------------------
// interactionModule: edge message (exp(logc)*sin(theta[src]-theta[dst])),
// segment-mean by dst, plus u0*[cos,sin](theta).  Output [N,3] = [w | vx | vy].
//
// MI455X reasoning:
//  - HBM traffic: src+dst = 256MB once (~11us @ 23.3TB/s)  -> not the limit.
//  - theta (2MB) and accumulator (4MB) live in the 192MB L2 -> limit is L2
//    transactions: 64M random gathers + scatter atomics.
//  - Halve atomic count: ONE u64 atomic per edge packs {count:[63:44],
//    Q32.32 fixed-point msg sum:[43:0]}.  Integer adds => exact AND
//    order-independent (deterministic across graph replays).
//  - CDNA5 path: async global->LDS double-buffered streaming of src/dst with
//    th:TH_LOAD_NT (protect L2 residency of theta/acc), s_wait_asynccnt.
// ---------------------------------------------------------------------------

#define WG    256
#define EPT   4                 // edges per thread per chunk (one b128 each)
#define CHUNK (WG * EPT)        // 1024 edges per block-iteration

__device__ __forceinline__ uint32_t lds_lo32(const void* p) {
  // generic pointer to LDS: low 32 bits are the wave-relative LDS byte offset
  return (uint32_t)(uintptr_t)p;
}

__device__ __forceinline__ void async_b128_to_lds(uint32_t lds_off,
                                                  const int* sbase,
                                                  uint32_t voff_bytes) {
  // GVS mode: copies 16 bytes from SGPR-pair base sbase plus VGPR byte offset voff_bytes to LDS at lds_off
  asm volatile("global_load_async_to_lds_b128 %0, %1, %2 th:TH_LOAD_NT"
               :: "v"(lds_off), "v"(voff_bytes), "s"(sbase)
               : "memory");
}

__global__ void zero_acc_kernel(unsigned long long* __restrict__ acc, int n) {
  int i = blockIdx.x * blockDim.x + threadIdx.x;
  if (i < n) acc[i] = 0ULL;
}

__global__ void __launch_bounds__(WG)
edge_kernel(const float* __restrict__ theta,
            const float* __restrict__ logc,
            const int* __restrict__ src,
            const int* __restrict__ dst,
            unsigned long long* __restrict__ acc,
            int nEdges, int nChunks) {
  __shared__ __align__(16) int ls[2][CHUNK];
  __shared__ __align__(16) int ld[2][CHUNK];

  const int tid = threadIdx.x;
  const float cexp = expf(logc[0]);

  auto issue = [&](uint32_t chunk, int b) {
    uint32_t ebase = chunk * CHUNK + tid * EPT;      // element index
    uint32_t boff  = ebase * 4u;                     // byte offset (<128MB)
    async_b128_to_lds(lds_lo32(&ls[b][tid * EPT]), src, boff);
    async_b128_to_lds(lds_lo32(&ld[b][tid * EPT]), dst, boff);
  };

  auto do_edge = [&](int s, int d) {
    float dth = theta[s] - theta[d];
    float msg = cexp * sinf(dth);
    long long q = __float2ll_rn(msg * 4294967296.0f);         // Q32.32
    unsigned long long pack = (1ULL << 44) + (unsigned long long)q;
    atomicAdd(&acc[d], pack);                                  // u64, no-return
  };

  uint32_t chunk  = blockIdx.x;
  uint32_t stride = gridDim.x;
  int  b = 0;
  bool curFull = false;

  if (chunk < (uint32_t)nChunks) {
    curFull = ((chunk + 1) * CHUNK <= (uint32_t)nEdges);
    if (curFull) issue(chunk, 0);
  }

  for (; chunk < (uint32_t)nChunks; chunk += stride) {
    uint32_t next = chunk + stride;
    bool nextFull = false;
    if (next < (uint32_t)nChunks &&
        ((next + 1) * CHUNK <= (uint32_t)nEdges)) {
      nextFull = true;
      issue(next, b ^ 1);
      // 2 ops outstanding for 'next'; everything older (current buffer) done.
      asm volatile("s_wait_asynccnt 0x2" ::: "memory");
    } else {
      asm volatile("s_wait_asynccnt 0x0" ::: "memory");
    }

    if (curFull) {
      int4 sv = *(const int4*)&ls[b][tid * EPT];     // ds_load_b128
      int4 dv = *(const int4*)&ld[b][tid * EPT];
      do_edge(sv.x, dv.x);
      do_edge(sv.y, dv.y);
      do_edge(sv.z, dv.z);
      do_edge(sv.w, dv.w);
    } else {
      uint32_t ebase = chunk * CHUNK + tid * EPT;
      #pragma unroll
      for (int j = 0; j < EPT; ++j) {
        uint32_t e = ebase + j;
        if (e < (uint32_t)nEdges) do_edge(src[e], dst[e]);
      }
    }
    b ^= 1;
    curFull = nextFull;
  }
}

__global__ void node_kernel(const float* __restrict__ theta,
                            const float* __restrict__ u0p,
                            const unsigned long long* __restrict__ acc,
                            float* __restrict__ out, int n) {
  int i = blockIdx.x * blockDim.x + threadIdx.x;
  if (i >= n) return;
  long long t   = (long long)acc[i];
  long long cnt = (t + (1LL << 43)) >> 44;           // round-to-nearest decode
  long long fix = t - (cnt << 44);                   // signed Q32.32 sum
  float w = 0.0f;
  if (cnt > 0)
    w = (float)((double)fix * (1.0 / 4294967296.0)) / (float)cnt;
  float th = theta[i], u0 = u0p[0];
  float s, c;
  sincosf(th, &s, &c);
  out[3 * i + 0] = w;
  out[3 * i + 1] = u0 * c;
  out[3 * i + 2] = u0 * s;
}

extern "C" void kernel_launch(void* const* d_in, const int* in_sizes, int n_in,
                              void* d_out, int out_size, void* d_ws, size_t ws_size,
                              hipStream_t stream) {
  const float* theta = (const float*)d_in[0];
  const float* logc  = (const float*)d_in[1];
  const float* u0    = (const float*)d_in[2];
  const int*   src   = (const int*)d_in[3];
  const int*   dst   = (const int*)d_in[4];
  float* out = (float*)d_out;

  int N = in_sizes[0];            // 500,000
  int E = in_sizes[3];            // 32,000,000

  unsigned long long* acc = (unsigned long long*)d_ws;  // N * 8B = 4MB

  zero_acc_kernel<<<(N + 255) / 256, 256, 0, stream>>>(acc, N);

  int nChunks = (E + CHUNK - 1) / CHUNK;                // 31,250 (exact)
  int blocks  = nChunks < 2048 ? nChunks : 2048;        // persistent blocks
  edge_kernel<<<blocks, WG, 0, stream>>>(theta, logc, src, dst, acc, E, nChunks);

  node_kernel<<<(N + 255) / 256, 256, 0, stream>>>(theta, u0, acc, out, N);
}